// KeypointBatchToGT_53008486367484
// MI455X (gfx1250) — compile-verified
//
#include <hip/hip_runtime.h>

// KeypointBatchToGT for MI455X (gfx1250, wave32).
// Layout (floats) in d_out, concatenated in reference return order:
//   [0,        2N)  gt_xy      = min(xy, 0.63)            (N = B*K keypoints)
//   [2N,       3N)  gt_loc_z   = min(z, 10.0)
//   [3N,       6N)  gt_index_z = { b_id, round(x/.01), round(y/.01) } per kp
//
// Each thread handles 4 consecutive keypoints:
//   in : 3 x global_load_b128   (48 bytes, 16B aligned)
//   out: 2 + 1 + 3 = 6 x global_store_b128 (96 bytes, all 16B aligned)

#define LOC_DELTA 0.01f
#define MAX_LOC   0.63f     // (64 - 1) * 0.01
#define MAX_VAL_Z 10.0f
#define K_LOG2    8         // K = 256 keypoints per batch row

__global__ __launch_bounds__(256) void
kp_batch_to_gt(const float4* __restrict__ in4,   // B*K*3 floats viewed as float4
               float* __restrict__ out,
               int n_quads)                      // (B*K)/4
{
    const int tid = blockIdx.x * blockDim.x + threadIdx.x;
    if (tid >= n_quads) return;
    const long long t = tid;
    const long long n_kp = (long long)n_quads * 4;

    // Speculative prefetch of input ~16KB ahead (gfx1250 global_prefetch_b8;
    // past-the-end addresses are silently dropped).
    __builtin_prefetch((const char*)(in4 + 3 * t) + 16384, 0, 3);

    // ---- load 4 keypoints: {x0,y0,z0,x1} {y1,z1,x2,y2} {z2,x3,y3,z3} ----
    const float4 a = in4[3 * t + 0];
    const float4 b = in4[3 * t + 1];
    const float4 c = in4[3 * t + 2];

    const float x0 = a.x, y0 = a.y, z0 = a.z, x1 = a.w;
    const float y1 = b.x, z1 = b.y, x2 = b.z, y2 = b.w;
    const float z2 = c.x, x3 = c.y, y3 = c.z, z3 = c.w;

    // ---- clamp xy to grid extent ----
    const float mx0 = fminf(x0, MAX_LOC), my0 = fminf(y0, MAX_LOC);
    const float mx1 = fminf(x1, MAX_LOC), my1 = fminf(y1, MAX_LOC);
    const float mx2 = fminf(x2, MAX_LOC), my2 = fminf(y2, MAX_LOC);
    const float mx3 = fminf(x3, MAX_LOC), my3 = fminf(y3, MAX_LOC);

    // ---- LocationToIndex: IEEE divide + round-to-nearest-even (== jnp.round) ----
    const float ix0 = rintf(mx0 / LOC_DELTA), iy0 = rintf(my0 / LOC_DELTA);
    const float ix1 = rintf(mx1 / LOC_DELTA), iy1 = rintf(my1 / LOC_DELTA);
    const float ix2 = rintf(mx2 / LOC_DELTA), iy2 = rintf(my2 / LOC_DELTA);
    const float ix3 = rintf(mx3 / LOC_DELTA), iy3 = rintf(my3 / LOC_DELTA);

    // ---- gt_xy : 8 floats per thread, two b128 stores ----
    float4* __restrict__ gt_xy = (float4*)out;
    gt_xy[2 * t + 0] = make_float4(mx0, my0, mx1, my1);
    gt_xy[2 * t + 1] = make_float4(mx2, my2, mx3, my3);

    // ---- gt_loc_z : 4 floats per thread, one b128 store ----
    float4* __restrict__ gt_z = (float4*)(out + 2 * n_kp);
    gt_z[t] = make_float4(fminf(z0, MAX_VAL_Z), fminf(z1, MAX_VAL_Z),
                          fminf(z2, MAX_VAL_Z), fminf(z3, MAX_VAL_Z));

    // ---- gt_index_z : 12 floats per thread, three b128 stores ----
    // 4 consecutive keypoints always share one batch row (4 | K=256):
    //   b_id = (4*tid) >> K_LOG2 = tid >> (K_LOG2 - 2)
    const float bf = (float)(tid >> (K_LOG2 - 2));
    float4* __restrict__ gt_idx = (float4*)(out + 3 * n_kp);
    gt_idx[3 * t + 0] = make_float4(bf,  ix0, iy0, bf );
    gt_idx[3 * t + 1] = make_float4(ix1, iy1, bf,  ix2);
    gt_idx[3 * t + 2] = make_float4(iy2, bf,  ix3, iy3);
}

extern "C" void kernel_launch(void* const* d_in, const int* in_sizes, int n_in,
                              void* d_out, int out_size, void* d_ws, size_t ws_size,
                              hipStream_t stream)
{
    (void)n_in; (void)out_size; (void)d_ws; (void)ws_size;

    const float4* in4 = (const float4*)d_in[0];
    float* out = (float*)d_out;

    const int n_kp    = in_sizes[0] / 3;   // B*K = 4,194,304
    const int n_quads = n_kp / 4;          // 1,048,576 threads (wave32: 8 waves/block)

    const int block = 256;
    const int grid  = (n_quads + block - 1) / block;   // 4096 blocks

    kp_batch_to_gt<<<grid, block, 0, stream>>>(in4, out, n_quads);
}